// RNN_67284957659813
// MI455X (gfx1250) — compile-verified
//
#include <hip/hip_runtime.h>
#include <cmath>
#include <cstdint>

// LSTM over (B=64, T=512, F=1024, H=1024) for MI455X (gfx1250).
//  - fp32 WMMA (v_wmma_f32_16x16x4_f32) for both GEMMs (fp32 accuracy across scan).
//  - Wi/Wh repacked once into lane-packed tiles -> B operand = 1 global_load_b128
//    per gate per 8-K instead of 4x global_load_b32 (4x fewer issue slots,
//    512B-contiguous streams through L2).
//  - A panels staged to LDS with gfx1250 async copies (global_load_async_to_lds_b64
//    via inline asm, ASYNCcnt + s_wait_asynccnt builtin) in a rotated swizzle ->
//    conflict-free ds_load_b64 A fetches.
//  - x_proj precomputed to ws (T,B,4H); recurrence = 512 launches, h double-buffered.

constexpr int Bsz = 64;
constexpr int Tsz = 512;
constexpr int Fsz = 1024;
constexpr int Hsz = 1024;
constexpr int Gsz = 4 * Hsz;      // 4096 gate columns
constexpr int MROWS = Bsz * Tsz;  // 32768 rows of the projection GEMM
constexpr int KROWS = 1024;       // K extent of both weights (F == H)
constexpr int PW_GROUP = (KROWS / 8) * 32 * 4;  // floats per 16-col group = 16384

typedef __attribute__((ext_vector_type(2))) float v2f;
typedef __attribute__((ext_vector_type(8))) float v8f;

#define DEVFN __device__ __forceinline__

#if defined(__has_builtin)
#if __has_builtin(__builtin_amdgcn_wmma_f32_16x16x4_f32)
#define HAVE_WMMA_F32X4 1
#endif
#if __has_builtin(__builtin_amdgcn_s_wait_asynccnt)
#define WAIT_ASYNC_BUILTIN 1
#endif
#endif

DEVFN v8f wmma16x16x4(v2f a, v2f b, v8f c) {
#if defined(HAVE_WMMA_F32X4)
  // (neg_a, A, neg_b, B, c_mod, C, reuse_a, reuse_b)
  return __builtin_amdgcn_wmma_f32_16x16x4_f32(false, a, false, b, (short)0, c,
                                               false, false);
#else
  // COMPILE-PROBE FALLBACK ONLY (histogram would show wmma==0).
  c[0] += a[0] * b[0] + a[1] * b[1];
  return c;
#endif
}

DEVFN v2f mkv2(float x, float y) {
  v2f r;
  r.x = x;
  r.y = y;
  return r;
}

// ---- gfx1250 async global->LDS copy (8 bytes), tracked by ASYNCcnt ---------
// Inline asm: VDST = VGPR holding the LDS byte address (flat addr[31:0]),
// VADDR = 64-bit global address pair, no SADDR ("off").
DEVFN void async_copy8(const float* g, const float* l) {
  const unsigned long long ga = (unsigned long long)(uintptr_t)g;
  const unsigned int la = (unsigned int)(uintptr_t)l;  // flat LDS addr[31:0]
  asm volatile("global_load_async_to_lds_b64 %0, %1, off" ::"v"(la), "v"(ga)
               : "memory");
}

DEVFN void wait_async0() {
#if defined(WAIT_ASYNC_BUILTIN)
  __builtin_amdgcn_s_wait_asynccnt(0);
#else
  asm volatile("s_wait_asynccnt 0" ::: "memory");
#endif
}

// Stage a 16-row x 1024-col fp32 panel (rows `rstride` floats apart) into LDS
// in the per-lane WMMA-A layout, via async copies:
//   chunk c (= k/4) holds 64 floats; lane L's v2f lives at
//   lds[c*64 + ((L + c) & 31)*2]  (rotation -> conflict-free ds_load_b64).
//   Lanes 0-15 hold rows 0-15, K = {4c,4c+1}; lanes 16-31 hold K = {4c+2,4c+3}.
// Caller must wait_async0() + __syncthreads(). Requires blockDim.x == 256.
DEVFN void stage_panel16xK(const float* __restrict__ src, size_t rstride,
                           float* __restrict__ lds) {
  const int m = threadIdx.x >> 4;   // row 0..15
  const int q0 = threadIdx.x & 15;  // chunk phase (256B coalesced per row)
  const float* row = src + (size_t)m * rstride;
#pragma unroll
  for (int i = 0; i < 16; ++i) {
    const int q = i * 16 + q0;  // chunk id == k0>>2
    const int slo = (m + q) & 31;
    const int shi = (m + 16 + q) & 31;
    float* d = lds + q * 64;
    async_copy8(row + q * 4, d + slo * 2);      // K = 4q, 4q+1
    async_copy8(row + q * 4 + 2, d + shi * 2);  // K = 4q+2, 4q+3
  }
}

DEVFN v2f lds_a_fetch(const float* __restrict__ lds, int k0, int lane) {
  const int ch = k0 >> 2;
  return *(const v2f*)(lds + ch * 64 + ((lane + ch) & 31) * 2);
}

DEVFN float sigmoidf_(float v) { return 1.0f / (1.0f + expf(-v)); }

// ---------------------------------------------------------------------------
// Weight repack: W (1024 x 4096 row-major) -> lane-packed b128 tiles.
// PW float index ((g*128 + p)*32 + l)*4 + j  holds
//   W[(p*8 + (l>>4)*2 + (j>>1)*4 + (j&1)) * Gsz + g*16 + (l&15)]
// so in the K-loop each lane's B operand for a K-pair is one float4.
// ---------------------------------------------------------------------------
__global__ __launch_bounds__(256) void repack_weights_kernel(
    const float* __restrict__ W, float* __restrict__ PW) {
  const int idx = blockIdx.x * blockDim.x + threadIdx.x;  // 256*128*32 total
  const int l = idx & 31;
  const int p = (idx >> 5) & 127;
  const int g = idx >> 12;  // 0..255
  if (g >= Gsz / 16) return;
  const int col = g * 16 + (l & 15);
  const int r0 = p * 8 + ((l >> 4) << 1);
  float4 v;
  v.x = W[(size_t)(r0 + 0) * Gsz + col];
  v.y = W[(size_t)(r0 + 1) * Gsz + col];
  v.z = W[(size_t)(r0 + 4) * Gsz + col];
  v.w = W[(size_t)(r0 + 5) * Gsz + col];
  *(float4*)(PW + (size_t)idx * 4) = v;
}

// ---------------------------------------------------------------------------
// Kernel 1: x_proj[(t*B + b), g] = sum_f x[b,t,f] * Wi[f,g]   (packed Wi)
// Grid: (MROWS/16) m-tiles x 8 blocks; 8 waves/block; wave -> 16x64 output.
// ---------------------------------------------------------------------------
__global__ __launch_bounds__(256) void xproj_kernel(
    const float* __restrict__ x, const float* __restrict__ pwi,
    float* __restrict__ xp) {
  __shared__ float ldsA[16 * Fsz];  // 64 KB swizzled A panel
  const int lane = threadIdx.x & 31;
  const int wave = threadIdx.x >> 5;
  const int mt = blockIdx.x >> 3;  // 0..2047 m-tile
  const int nsg = blockIdx.x & 7;  // n-strip group
  const int l16 = lane & 15;

  {
    // Rows r0..r0+15 of the (t*B + b) ordering: constant stride T*F in x.
    const int r0 = mt * 16;
    const int b0 = r0 & (Bsz - 1);
    const int t0 = r0 >> 6;  // r = t*64 + b
    stage_panel16xK(x + ((size_t)b0 * Tsz + t0) * Fsz, (size_t)Tsz * Fsz, ldsA);
  }
  wait_async0();
  __syncthreads();

  const int ns = nsg * 8 + wave;  // 0..63
  const int n0 = ns * 64;
  const float* b0p = pwi + (size_t)(ns * 4) * PW_GROUP + lane * 4;
  const float* b1p = b0p + PW_GROUP;
  const float* b2p = b0p + 2 * PW_GROUP;
  const float* b3p = b0p + 3 * PW_GROUP;

  v8f acc0 = {}, acc1 = {}, acc2 = {}, acc3 = {};
#pragma unroll 4
  for (int p = 0; p < 128; ++p) {
    const int k0 = p * 8;
    const v2f a0 = lds_a_fetch(ldsA, k0, lane);
    const v2f a1 = lds_a_fetch(ldsA, k0 + 4, lane);
    __builtin_prefetch(b0p + (size_t)(p + 16) * 128, 0, 0);
    const float4 v0 = *(const float4*)(b0p + (size_t)p * 128);
    const float4 v1 = *(const float4*)(b1p + (size_t)p * 128);
    const float4 v2 = *(const float4*)(b2p + (size_t)p * 128);
    const float4 v3 = *(const float4*)(b3p + (size_t)p * 128);
    acc0 = wmma16x16x4(a0, mkv2(v0.x, v0.y), acc0);
    acc1 = wmma16x16x4(a0, mkv2(v1.x, v1.y), acc1);
    acc2 = wmma16x16x4(a0, mkv2(v2.x, v2.y), acc2);
    acc3 = wmma16x16x4(a0, mkv2(v3.x, v3.y), acc3);
    acc0 = wmma16x16x4(a1, mkv2(v0.z, v0.w), acc0);
    acc1 = wmma16x16x4(a1, mkv2(v1.z, v1.w), acc1);
    acc2 = wmma16x16x4(a1, mkv2(v2.z, v2.w), acc2);
    acc3 = wmma16x16x4(a1, mkv2(v3.z, v3.w), acc3);
  }

  // C/D layout: VGPR j -> row j (lanes 0-15) / row j+8 (lanes 16-31)
  const int mbase = mt * 16 + (lane >> 4) * 8;
#pragma unroll
  for (int j = 0; j < 8; ++j) {
    float* o = xp + (size_t)(mbase + j) * Gsz + n0 + l16;
    o[0] = acc0[j];
    o[16] = acc1[j];
    o[32] = acc2[j];
    o[48] = acc3[j];
  }
}

// ---------------------------------------------------------------------------
// Kernel 2: one LSTM time step (packed Wh).
// z = xp[t] + h_in @ Wh + b ; gates ; out[:,t,:] = new_h ;
// carry-select on t < lengths[b] (h double-buffered, c in place).
// Grid: 32 blocks x 256 threads. Wave -> 16 batch rows x 16 H-cols x 4 gates.
// ---------------------------------------------------------------------------
__global__ __launch_bounds__(256) void lstm_step_kernel(
    const float* __restrict__ xp, const float* __restrict__ pwh,
    const float* __restrict__ bias, const int* __restrict__ lengths,
    float* __restrict__ c_state, const float* __restrict__ h_in,
    float* __restrict__ h_out, float* __restrict__ out, int t) {
  __shared__ float ldsA[16 * Hsz];  // 64 KB swizzled h panel
  const int lane = threadIdx.x & 31;
  const int wave = threadIdx.x >> 5;
  const int m0 = (blockIdx.x >> 3) * 16;       // batch tile base
  const int nh = (blockIdx.x & 7) * 8 + wave;  // 0..63 -> 16 H-cols
  const int l16 = lane & 15;
  const int hi8 = (lane >> 4) * 8;
  const int hcol = nh * 16 + l16;

  stage_panel16xK(h_in + (size_t)m0 * Hsz, (size_t)Hsz, ldsA);

  // Accumulators start at xp[t] + bias (overlaps with the async staging).
  v8f zi, zf, zg, zo;
  {
    const float bi = bias[hcol];
    const float bf = bias[Hsz + hcol];
    const float bg = bias[2 * Hsz + hcol];
    const float bo = bias[3 * Hsz + hcol];
    const float* xpt = xp + ((size_t)t * Bsz + m0 + hi8) * Gsz + hcol;
#pragma unroll
    for (int j = 0; j < 8; ++j) {
      const float* zr = xpt + (size_t)j * Gsz;
      zi[j] = zr[0] + bi;
      zf[j] = zr[Hsz] + bf;
      zg[j] = zr[2 * Hsz] + bg;
      zo[j] = zr[3 * Hsz] + bo;
    }
  }

  wait_async0();
  __syncthreads();

  // Gate gi columns live at gi*1024 + nh*16 -> packed group gi*64 + nh.
  const float* pI = pwh + (size_t)(0 * 64 + nh) * PW_GROUP + lane * 4;
  const float* pF = pwh + (size_t)(1 * 64 + nh) * PW_GROUP + lane * 4;
  const float* pG = pwh + (size_t)(2 * 64 + nh) * PW_GROUP + lane * 4;
  const float* pO = pwh + (size_t)(3 * 64 + nh) * PW_GROUP + lane * 4;

#pragma unroll 4
  for (int p = 0; p < 128; ++p) {
    const int k0 = p * 8;
    const v2f a0 = lds_a_fetch(ldsA, k0, lane);
    const v2f a1 = lds_a_fetch(ldsA, k0 + 4, lane);
    __builtin_prefetch(pI + (size_t)(p + 16) * 128, 0, 0);
    const float4 vI = *(const float4*)(pI + (size_t)p * 128);
    const float4 vF = *(const float4*)(pF + (size_t)p * 128);
    const float4 vG = *(const float4*)(pG + (size_t)p * 128);
    const float4 vO = *(const float4*)(pO + (size_t)p * 128);
    zi = wmma16x16x4(a0, mkv2(vI.x, vI.y), zi);
    zf = wmma16x16x4(a0, mkv2(vF.x, vF.y), zf);
    zg = wmma16x16x4(a0, mkv2(vG.x, vG.y), zg);
    zo = wmma16x16x4(a0, mkv2(vO.x, vO.y), zo);
    zi = wmma16x16x4(a1, mkv2(vI.z, vI.w), zi);
    zf = wmma16x16x4(a1, mkv2(vF.z, vF.w), zf);
    zg = wmma16x16x4(a1, mkv2(vG.z, vG.w), zg);
    zo = wmma16x16x4(a1, mkv2(vO.z, vO.w), zo);
  }

#pragma unroll
  for (int j = 0; j < 8; ++j) {
    const int m = m0 + hi8 + j;  // batch index owned by this lane/register
    const float ii = sigmoidf_(zi[j]);
    const float ff = sigmoidf_(zf[j]);
    const float gg = tanhf(zg[j]);
    const float oo = sigmoidf_(zo[j]);
    const size_t sidx = (size_t)m * Hsz + hcol;
    const float c_old = c_state[sidx];
    // previous h for this element, re-read from the staged (swizzled) panel
    const int ch = hcol >> 2;
    const int lanep = ((hcol & 2) ? 16 : 0) + hi8 + j;
    const float h_old = ldsA[ch * 64 + ((lanep + ch) & 31) * 2 + (hcol & 1)];
    const float nc = ff * c_old + ii * gg;
    const float nhv = oo * tanhf(nc);
    out[((size_t)m * Tsz + t) * Hsz + hcol] = nhv;  // ys = new_h (unmasked)
    const bool upd = t < lengths[m];                // carry-select
    c_state[sidx] = upd ? nc : c_old;
    h_out[sidx] = upd ? nhv : h_old;
  }
}

__global__ void zero_kernel(float* __restrict__ p, int n) {
  const int i = blockIdx.x * blockDim.x + threadIdx.x;
  if (i < n) p[i] = 0.0f;
}

extern "C" void kernel_launch(void* const* d_in, const int* in_sizes, int n_in,
                              void* d_out, int out_size, void* d_ws,
                              size_t ws_size, hipStream_t stream) {
  const float* x = (const float*)d_in[0];
  const float* Wi = (const float*)d_in[1];
  const float* Wh = (const float*)d_in[2];
  const float* bias = (const float*)d_in[3];
  const int* lengths = (const int*)d_in[4];
  float* out = (float*)d_out;

  // Workspace: xp (T,B,4H) | h0 | h1 | c | packed Wi | packed Wh  (~545 MB)
  float* xp = (float*)d_ws;
  const size_t xp_elems = (size_t)Tsz * Bsz * Gsz;
  float* h0 = xp + xp_elems;
  float* h1 = h0 + (size_t)Bsz * Hsz;
  float* cs = h1 + (size_t)Bsz * Hsz;
  float* pwi = cs + (size_t)Bsz * Hsz;
  float* pwh = pwi + (size_t)KROWS * Gsz;

  const int state_elems = 3 * Bsz * Hsz;  // h0,h1,c contiguous
  zero_kernel<<<(state_elems + 255) / 256, 256, 0, stream>>>(h0, state_elems);

  const int repack_blocks = (Gsz / 16) * (KROWS / 8) * 32 / 256;  // 4096
  repack_weights_kernel<<<repack_blocks, 256, 0, stream>>>(Wi, pwi);
  repack_weights_kernel<<<repack_blocks, 256, 0, stream>>>(Wh, pwh);

  xproj_kernel<<<(MROWS / 16) * 8, 256, 0, stream>>>(x, pwi, xp);

  for (int t = 0; t < Tsz; ++t) {
    const float* hin = (t & 1) ? h1 : h0;
    float* hout = (t & 1) ? h0 : h1;
    lstm_step_kernel<<<32, 256, 0, stream>>>(xp, pwh, bias, lengths, cs, hin,
                                             hout, out, t);
  }
  (void)in_sizes;
  (void)n_in;
  (void)out_size;
  (void)ws_size;
}